// RecGNN_22677427322907
// MI455X (gfx1250) — compile-verified
//
#include <hip/hip_runtime.h>
#include <hip/hip_bf16.h>

#define N_NODES 200000
#define E_EDGES 600000
#define DHID 128
#define DXF 3
#define DMLP 256
#define NLEV 16

typedef __attribute__((ext_vector_type(2))) float v2f;
typedef __attribute__((ext_vector_type(8))) float v8f;

// meta layout (ints): [0..15] edge counts, [16..31] node counts,
// [32..48] edge offsets (17), [49..65] node offsets (17),
// [66..81] edge cursors, [82..97] node cursors
#define M_EC 0
#define M_NC 16
#define M_EOFF 32
#define M_NOFF 49
#define M_ECUR 66
#define M_NCUR 82

__device__ __forceinline__ v8f wmma4(v2f a, v2f b, v8f c) {
  // V_WMMA_F32_16X16X4_F32 : D = A(16x4 f32) x B(4x16 f32) + C(16x16 f32)
  return __builtin_amdgcn_wmma_f32_16x16x4_f32(
      /*neg_a=*/false, a, /*neg_b=*/false, b,
      /*c_mod=*/(short)0, c, /*reuse_a=*/false, /*reuse_b=*/false);
}

// out[16 x 16] tile of  A(16xK, LDS row-major, stride LDA)  @  W^T
// where W is row-major [cols x LDB] and B[k][n] = W[col0+n][k].
template <int LDA, int LDB, int K>
__device__ __forceinline__ v8f gemm16(const float* __restrict__ Alds,
                                      const float* __restrict__ W,
                                      int col0, v8f acc) {
  const int lane = threadIdx.x & 31;
  const int n = lane & 15;          // B/D column within tile, A row (M)
  const int kh = (lane >> 4) << 1;  // K sub-offset: lanes 16..31 take K+2,K+3
  const float* arow = Alds + n * LDA + kh;
  const float* wrow = W + (size_t)(col0 + n) * LDB + kh;
#pragma unroll
  for (int k0 = 0; k0 < K; k0 += 4) {
    v2f a = *(const v2f*)(arow + k0);
    v2f b = *(const v2f*)(wrow + k0);
    acc = wmma4(a, b, acc);
  }
  return acc;
}

__device__ __forceinline__ float sigmoidf_(float x) {
  return 1.0f / (1.0f + __expf(-x));
}

// ---------------- init: h = W_emd + b_emd broadcast, msg = 0, meta = 0 -----
__global__ void k_init(float* __restrict__ h, float* __restrict__ msg,
                       const float* __restrict__ W_emd,
                       const float* __restrict__ b_emd, int* __restrict__ meta) {
  int i = blockIdx.x * 256 + threadIdx.x;
  if (i < N_NODES * DHID) {
    int d = i & (DHID - 1);
    h[i] = W_emd[d] + b_emd[d];
    msg[i] = 0.0f;
  }
  if (blockIdx.x == 0 && threadIdx.x < 128) meta[threadIdx.x] = 0;
}

// ---------------- histogram of edge dst-levels and node levels -------------
__global__ void k_hist(const int* __restrict__ edst,
                       const int* __restrict__ flevel, int* __restrict__ meta) {
  int i = blockIdx.x * 256 + threadIdx.x;
  if (i < E_EDGES) atomicAdd(&meta[M_EC + flevel[edst[i]]], 1);
  if (i < N_NODES) atomicAdd(&meta[M_NC + flevel[i]], 1);
}

// ---------------- tiny exclusive scan (16 bins) ----------------------------
__global__ void k_scan(int* __restrict__ meta) {
  int eo = 0, no = 0;
#pragma unroll
  for (int l = 0; l < NLEV; l++) {
    meta[M_EOFF + l] = eo;
    meta[M_ECUR + l] = eo;
    eo += meta[M_EC + l];
    meta[M_NOFF + l] = no;
    meta[M_NCUR + l] = no;
    no += meta[M_NC + l];
  }
  meta[M_EOFF + NLEV] = eo;
  meta[M_NOFF + NLEV] = no;
}

// ---------------- counting-sort scatter ------------------------------------
__global__ void k_scatter(const int* __restrict__ esrc,
                          const int* __restrict__ edst,
                          const int* __restrict__ flevel,
                          int* __restrict__ es_src, int* __restrict__ es_dst,
                          int* __restrict__ node_sorted,
                          int* __restrict__ meta) {
  int i = blockIdx.x * 256 + threadIdx.x;
  if (i < E_EDGES) {
    int d = edst[i];
    int pos = atomicAdd(&meta[M_ECUR + flevel[d]], 1);
    es_src[pos] = esrc[i];
    es_dst[pos] = d;
  }
  if (i < N_NODES) {
    int pos = atomicAdd(&meta[M_NCUR + flevel[i]], 1);
    node_sorted[pos] = i;
  }
}

// ---------------- per-edge message GEMM + scatter-add ----------------------
// tile = 16 edges; A = gathered h[src] (16x128); out = A @ W_aggr^T + b_aggr
__global__ void __launch_bounds__(256) k_msg(
    const float* __restrict__ h, float* __restrict__ msg,
    const int* __restrict__ es_src, const int* __restrict__ es_dst,
    const int* __restrict__ meta, const float* __restrict__ W_aggr,
    const float* __restrict__ b_aggr, int level) {
  __shared__ float sA[16 * DHID];
  __shared__ int sDst[16];
  const int beg = meta[M_EOFF + level];
  const int end = meta[M_EOFF + level + 1];
  const int ntiles = (end - beg + 15) >> 4;
  const int t = threadIdx.x;
  const int lane = t & 31, wave = t >> 5;
  const int n = lane & 15, hf = lane >> 4;

  for (int tile = blockIdx.x; tile < ntiles; tile += gridDim.x) {
    const int base = beg + tile * 16;
    // row r of the tile is edge (base + r)
    {
      int r = t >> 4, c0 = (t & 15) * 8;
      int idx = base + r;
      int s = -1, d = -1;
      if (idx < end) { s = es_src[idx]; d = es_dst[idx]; }
      if ((t & 15) == 0) sDst[r] = d;
      float4 z = {0.f, 0.f, 0.f, 0.f};
      float4 v0 = z, v1 = z;
      if (s >= 0) {
        const float4* p = (const float4*)(h + (size_t)s * DHID + c0);
        v0 = p[0];
        v1 = p[1];
      }
      *(float4*)&sA[r * DHID + c0] = v0;
      *(float4*)&sA[r * DHID + c0 + 4] = v1;
    }
    __syncthreads();

    const int col0 = wave * 16;
    v8f acc = {0.f, 0.f, 0.f, 0.f, 0.f, 0.f, 0.f, 0.f};
    acc = gemm16<DHID, DHID, DHID>(sA, W_aggr, col0, acc);
    const float bias = b_aggr[col0 + n];
#pragma unroll
    for (int v = 0; v < 8; v++) {
      int M = v + 8 * hf;
      int d = sDst[M];
      if (d >= 0)
        atomicAdd(&msg[(size_t)d * DHID + col0 + n], acc[v] + bias);
    }
    __syncthreads();
  }
}

// ---------------- GRU update for level-l nodes -----------------------------
__global__ void __launch_bounds__(256) k_gru(
    float* __restrict__ h, const float* __restrict__ msg,
    const float* __restrict__ x, const int* __restrict__ node_sorted,
    const int* __restrict__ meta, const float* __restrict__ w_ih,
    const float* __restrict__ w_hh, const float* __restrict__ b_ih,
    const float* __restrict__ b_hh, int level) {
  __shared__ float sIn[16 * 132];  // [msg(128) | x(3) | 0]
  __shared__ float sH[16 * DHID];
  __shared__ int sNode[16];
  const int beg = meta[M_NOFF + level];
  const int end = meta[M_NOFF + level + 1];
  const int ntiles = (end - beg + 15) >> 4;
  const int t = threadIdx.x;
  const int lane = t & 31, wave = t >> 5;
  const int n = lane & 15, hf = lane >> 4, kh = hf << 1;

  for (int tile = blockIdx.x; tile < ntiles; tile += gridDim.x) {
    {
      int r = t >> 4, c0 = (t & 15) * 8;
      int idx = beg + tile * 16 + r;
      int nd = (idx < end) ? node_sorted[idx] : -1;
      if ((t & 15) == 0) sNode[r] = nd;
      float4 z = {0.f, 0.f, 0.f, 0.f};
      float4 m0 = z, m1 = z, h0 = z, h1 = z;
      if (nd >= 0) {
        const float4* pm = (const float4*)(msg + (size_t)nd * DHID + c0);
        const float4* ph = (const float4*)(h + (size_t)nd * DHID + c0);
        m0 = pm[0]; m1 = pm[1];
        h0 = ph[0]; h1 = ph[1];
      }
      *(float4*)&sIn[r * 132 + c0] = m0;
      *(float4*)&sIn[r * 132 + c0 + 4] = m1;
      *(float4*)&sH[r * DHID + c0] = h0;
      *(float4*)&sH[r * DHID + c0 + 4] = h1;
      if ((t & 15) == 0) {
        float x0 = 0.f, x1 = 0.f, x2 = 0.f;
        if (nd >= 0) {
          x0 = x[nd * 3 + 0];
          x1 = x[nd * 3 + 1];
          x2 = x[nd * 3 + 2];
        }
        sIn[r * 132 + 128] = x0;
        sIn[r * 132 + 129] = x1;
        sIn[r * 132 + 130] = x2;
        sIn[r * 132 + 131] = 0.f;
      }
    }
    __syncthreads();

    const int cr = wave * 16;        // r-gate columns
    const int cz = 128 + wave * 16;  // z-gate columns
    const int cn = 256 + wave * 16;  // n-gate columns
    v8f gir = {0.f,0.f,0.f,0.f,0.f,0.f,0.f,0.f};
    v8f giz = gir, gin = gir, ghr = gir, ghz = gir, ghn = gir;

    gir = gemm16<132, 131, 128>(sIn, w_ih, cr, gir);
    giz = gemm16<132, 131, 128>(sIn, w_ih, cz, giz);
    gin = gemm16<132, 131, 128>(sIn, w_ih, cn, gin);
    // tail K-step for the 3 x-features (w_ih cols 128..130, col 131 == 0)
    {
      v2f a = *(const v2f*)(sIn + n * 132 + 128 + kh);
      const float* wr = w_ih + (size_t)(cr + n) * 131;
      v2f b;
      b.x = (hf == 0) ? wr[128] : wr[130];
      b.y = (hf == 0) ? wr[129] : 0.f;
      gir = wmma4(a, b, gir);
      const float* wz = w_ih + (size_t)(cz + n) * 131;
      b.x = (hf == 0) ? wz[128] : wz[130];
      b.y = (hf == 0) ? wz[129] : 0.f;
      giz = wmma4(a, b, giz);
      const float* wn = w_ih + (size_t)(cn + n) * 131;
      b.x = (hf == 0) ? wn[128] : wn[130];
      b.y = (hf == 0) ? wn[129] : 0.f;
      gin = wmma4(a, b, gin);
    }
    ghr = gemm16<DHID, DHID, DHID>(sH, w_hh, cr, ghr);
    ghz = gemm16<DHID, DHID, DHID>(sH, w_hh, cz, ghz);
    ghn = gemm16<DHID, DHID, DHID>(sH, w_hh, cn, ghn);

    const float bir = b_ih[cr + n], biz = b_ih[cz + n], bin = b_ih[cn + n];
    const float bhr = b_hh[cr + n], bhz = b_hh[cz + n], bhn = b_hh[cn + n];
#pragma unroll
    for (int v = 0; v < 8; v++) {
      int M = v + 8 * hf;
      float r = sigmoidf_((gir[v] + bir) + (ghr[v] + bhr));
      float z = sigmoidf_((giz[v] + biz) + (ghz[v] + bhz));
      float nn = tanhf((gin[v] + bin) + r * (ghn[v] + bhn));
      float hprev = sH[M * DHID + cr + n];
      float hnew = (1.f - z) * nn + z * hprev;
      int nd = sNode[M];
      if (nd >= 0) h[(size_t)nd * DHID + cr + n] = hnew;
    }
    __syncthreads();
  }
}

// ---------------- fused predictor MLP: 128 -> 256 -> 256 -> 1 --------------
__global__ void __launch_bounds__(256) k_mlp(
    const float* __restrict__ h, const float* __restrict__ W1,
    const float* __restrict__ b1, const float* __restrict__ W2,
    const float* __restrict__ b2, const float* __restrict__ W3,
    const float* __restrict__ b3, float* __restrict__ out) {
  __shared__ float sH[16 * DHID];
  __shared__ float sH1[16 * DMLP];
  __shared__ float sH2[16 * DMLP];
  const int t = threadIdx.x;
  const int lane = t & 31, wave = t >> 5;
  const int n = lane & 15, hf = lane >> 4;
  const int ntiles = N_NODES / 16;  // 12500, exact

  for (int tile = blockIdx.x; tile < ntiles; tile += gridDim.x) {
    const int node0 = tile * 16;
    {
      int r = t >> 4, c0 = (t & 15) * 8;
      const float4* p = (const float4*)(h + (size_t)(node0 + r) * DHID + c0);
      *(float4*)&sH[r * DHID + c0] = p[0];
      *(float4*)&sH[r * DHID + c0 + 4] = p[1];
    }
    __syncthreads();
    // layer 1: 16 col tiles, each wave does 2
#pragma unroll
    for (int cb = 0; cb < 2; cb++) {
      const int col0 = wave * 16 + cb * 128;
      v8f acc = {0.f,0.f,0.f,0.f,0.f,0.f,0.f,0.f};
      acc = gemm16<DHID, DHID, DHID>(sH, W1, col0, acc);
      const float bias = b1[col0 + n];
#pragma unroll
      for (int v = 0; v < 8; v++)
        sH1[(v + 8 * hf) * DMLP + col0 + n] = fmaxf(acc[v] + bias, 0.f);
    }
    __syncthreads();
    // layer 2
#pragma unroll
    for (int cb = 0; cb < 2; cb++) {
      const int col0 = wave * 16 + cb * 128;
      v8f acc = {0.f,0.f,0.f,0.f,0.f,0.f,0.f,0.f};
      acc = gemm16<DMLP, DMLP, DMLP>(sH1, W2, col0, acc);
      const float bias = b2[col0 + n];
#pragma unroll
      for (int v = 0; v < 8; v++)
        sH2[(v + 8 * hf) * DMLP + col0 + n] = fmaxf(acc[v] + bias, 0.f);
    }
    __syncthreads();
    // layer 3: 16x256 @ 256x1 dot products; partial per (row, 16-col segment)
    {
      int r = t >> 4, seg = t & 15;
      const float* row = sH2 + r * DMLP + seg * 16;
      const float* w3 = W3 + seg * 16;
      float p = 0.f;
#pragma unroll
      for (int j = 0; j < 16; j++) p += row[j] * w3[j];
      sH[r * 16 + seg] = p;  // reuse sH as reduction scratch
    }
    __syncthreads();
    if (t < 16) {
      float s = 0.f;
#pragma unroll
      for (int j = 0; j < 16; j++) s += sH[t * 16 + j];
      out[node0 + t] = s + b3[0];
    }
    __syncthreads();
  }
}

extern "C" void kernel_launch(void* const* d_in, const int* in_sizes, int n_in,
                              void* d_out, int out_size, void* d_ws,
                              size_t ws_size, hipStream_t stream) {
  (void)in_sizes; (void)n_in; (void)out_size; (void)ws_size;
  const float* x = (const float*)d_in[0];
  const int* edge_index = (const int*)d_in[1];
  const int* esrc = edge_index;
  const int* edst = edge_index + E_EDGES;
  const int* flevel = (const int*)d_in[2];
  const float* W_emd = (const float*)d_in[3];
  const float* b_emd = (const float*)d_in[4];
  const float* W_aggr = (const float*)d_in[5];
  const float* b_aggr = (const float*)d_in[6];
  const float* w_ih = (const float*)d_in[7];
  const float* w_hh = (const float*)d_in[8];
  const float* b_ih = (const float*)d_in[9];
  const float* b_hh = (const float*)d_in[10];
  const float* W1 = (const float*)d_in[11];
  const float* b1 = (const float*)d_in[12];
  const float* W2 = (const float*)d_in[13];
  const float* b2 = (const float*)d_in[14];
  const float* W3 = (const float*)d_in[15];
  const float* b3 = (const float*)d_in[16];
  float* out = (float*)d_out;

  // workspace carve (all 4B types, segments 16B-aligned by construction)
  float* h = (float*)d_ws;                        // N*D
  float* msg = h + (size_t)N_NODES * DHID;        // N*D
  int* es_src = (int*)(msg + (size_t)N_NODES * DHID);  // E
  int* es_dst = es_src + E_EDGES;                 // E
  int* node_sorted = es_dst + E_EDGES;            // N
  int* meta = node_sorted + N_NODES;              // 128

  k_init<<<(N_NODES * DHID + 255) / 256, 256, 0, stream>>>(h, msg, W_emd,
                                                           b_emd, meta);
  k_hist<<<(E_EDGES + 255) / 256, 256, 0, stream>>>(edst, flevel, meta);
  k_scan<<<1, 1, 0, stream>>>(meta);
  k_scatter<<<(E_EDGES + 255) / 256, 256, 0, stream>>>(
      esrc, edst, flevel, es_src, es_dst, node_sorted, meta);

  for (int l = 1; l < NLEV; l++) {
    k_msg<<<960, 256, 0, stream>>>(h, msg, es_src, es_dst, meta, W_aggr,
                                   b_aggr, l);
    k_gru<<<512, 256, 0, stream>>>(h, msg, x, node_sorted, meta, w_ih, w_hh,
                                   b_ih, b_hh, l);
  }
  k_mlp<<<2048, 256, 0, stream>>>(h, W1, b1, W2, b2, W3, b3, out);
}